// BayesianDeeperGCN_42623255446347
// MI455X (gfx1250) — compile-verified
//
#include <hip/hip_runtime.h>
#include <hip/hip_bf16.h>

// ---------------------------------------------------------------------------
// DeeperGCN (GENConv softmax aggregation, res+ blocks) forward for MI455X.
// Dense matmuls use V_WMMA_F32_16X16X4_F32 (full fp32, matches reference).
// Graph segment ops use float4-coalesced gathers + f32 global atomics.
// N/K are compile-time template params so all load addresses fold into
// 24-bit immediate offsets (no per-iteration 64-bit address VALU).
// ---------------------------------------------------------------------------

#define NN      50000
#define EE      800000
#define HH      128      // IN_CH == HID
#define HH2     256      // 2*HID
#define OUTC    64
#define NLAYERS 3
#define EPS_GEN 1e-7f
#define EPS_NRM 1e-5f

typedef __attribute__((ext_vector_type(2))) float v2f;
typedef __attribute__((ext_vector_type(8))) float v8f;

#define LDS_PAD 4

// ---------------------------------------------------------------------------
// C[M,N] (+)= A[M,K] @ B[K,N] + bias[N]      (M%16==0, N%16==0, K%4==0)
// grid = (M/16, ceil(N/128)), block = 256 threads (8 waves).
// Wave w computes the 16x16 tile at columns n0 = by*128 + w*16.
// A tile staged in LDS with +4 padding so the 16-lane column reads hit
// distinct banks. WMMA f32 VGPR layout per ISA 7.12.2:
//   A (16x4):  lane = hi*16+m,  a.x = A[m][k+2*hi],   a.y = A[m][k+2*hi+1]
//   B (4x16):  lane = hi*16+n,  b.x = B[k+2*hi][n],   b.y = B[k+2*hi+1][n]
//   C (16x16): vgpr r, lane = hi*16+n  ->  C[r+8*hi][n]
// ---------------------------------------------------------------------------
template <int ACC, int RELU, int N, int K>
__global__ __launch_bounds__(256) void gemm_wmma_f32(
    const float* __restrict__ A, const float* __restrict__ B,
    const float* __restrict__ bias, float* __restrict__ C) {
  __shared__ float ldsA[16 * (K + LDS_PAD)];
  const int tid = threadIdx.x;
  const int m0  = blockIdx.x * 16;
  constexpr int Kp = K + LDS_PAD;

  // cooperative float4 load of the 16xK A tile
#pragma unroll
  for (int idx = tid * 4; idx < 16 * K; idx += 256 * 4) {
    const int r = idx / K;
    const int c = idx - r * K;                     // multiple of 4
    const float4 v = *(const float4*)(A + (size_t)(m0 + r) * K + c);
    *(float4*)(&ldsA[r * Kp + c]) = v;
  }
  __syncthreads();

  const int wave = tid >> 5;
  const int lane = tid & 31;
  const int hi   = lane >> 4;                      // 0 | 1
  const int ln   = lane & 15;
  const int n0   = blockIdx.y * 128 + wave * 16;
  if (n0 >= N) return;                             // wave-uniform: EXEC stays all-1s

  __builtin_prefetch(B + n0, 0, 1);                // global_prefetch_b8 on B panel

  v8f acc = {};
  if (ACC) {
#pragma unroll
    for (int r = 0; r < 8; ++r)
      acc[r] = C[(size_t)(m0 + hi * 8 + r) * N + (n0 + ln)];
  }

  const float* arow = &ldsA[ln * Kp + hi * 2];
  const float* bcol = B + (size_t)(hi * 2) * N + n0 + ln;
#pragma unroll 8
  for (int k = 0; k < K; k += 4) {
    v2f a, b;
    a.x = arow[k];                                 // ds_load, imm offset
    a.y = arow[k + 1];
    b.x = bcol[(size_t)k * N];                     // global_load, imm offset
    b.y = bcol[(size_t)(k + 1) * N];
    acc = __builtin_amdgcn_wmma_f32_16x16x4_f32(
        /*neg_a=*/false, a, /*neg_b=*/false, b,
        /*c_mod=*/(short)0, acc, /*reuse_a=*/false, /*reuse_b=*/false);
  }

  const float bv = bias[n0 + ln];
#pragma unroll
  for (int r = 0; r < 8; ++r) {
    float v = acc[r] + bv;
    if (RELU) v = fmaxf(v, 0.f);
    C[(size_t)(m0 + hi * 8 + r) * N + (n0 + ln)] = v;
  }
}

// --------------------------- LayerNorm + ReLU ------------------------------
// one wave per node, lane owns 4 contiguous features (float4)
__global__ __launch_bounds__(256) void ln_relu_kernel(
    const float* __restrict__ h, const float* __restrict__ g,
    const float* __restrict__ b, float* __restrict__ t) {
  const int node = blockIdx.x * 8 + (threadIdx.x >> 5);
  const int lane = threadIdx.x & 31;
  if (node >= NN) return;
  const float4 v = *(const float4*)(h + (size_t)node * HH + lane * 4);
  float s = v.x + v.y + v.z + v.w;
#pragma unroll
  for (int off = 16; off > 0; off >>= 1) s += __shfl_xor(s, off, 32);
  const float mu = s * (1.f / HH);
  float4 d;
  d.x = v.x - mu; d.y = v.y - mu; d.z = v.z - mu; d.w = v.w - mu;
  float q = d.x * d.x + d.y * d.y + d.z * d.z + d.w * d.w;
#pragma unroll
  for (int off = 16; off > 0; off >>= 1) q += __shfl_xor(q, off, 32);
  const float rs = rsqrtf(q * (1.f / HH) + EPS_NRM);
  const float4 gg = *(const float4*)(g + lane * 4);
  const float4 bb = *(const float4*)(b + lane * 4);
  float4 o;
  o.x = fmaxf(d.x * rs * gg.x + bb.x, 0.f);
  o.y = fmaxf(d.y * rs * gg.y + bb.y, 0.f);
  o.z = fmaxf(d.z * rs * gg.z + bb.z, 0.f);
  o.w = fmaxf(d.w * rs * gg.w + bb.w, 0.f);
  *(float4*)(t + (size_t)node * HH + lane * 4) = o;
}

// ---------------------- edge pass 1: segment max ---------------------------
// msg = relu(t[src])+eps > 0 always, so positive-float atomicMax on the int
// bit pattern is order-preserving; init value 0 == 0.0f matches reference's
// isfinite(seg_max)->0 replacement for empty segments.
__global__ __launch_bounds__(256) void edge_max_kernel(
    const float* __restrict__ t, const int* __restrict__ src,
    const int* __restrict__ dst, int* __restrict__ maxb) {
  const int e = blockIdx.x * 8 + (threadIdx.x >> 5);
  const int lane = threadIdx.x & 31;
  if (e >= EE) return;
  const int s = src[e];
  const int d = dst[e];
  const float4 v = *(const float4*)(t + (size_t)s * HH + lane * 4);
  int* p = maxb + (size_t)d * HH + lane * 4;
  atomicMax(p + 0, __float_as_int(fmaxf(v.x, 0.f) + EPS_GEN));
  atomicMax(p + 1, __float_as_int(fmaxf(v.y, 0.f) + EPS_GEN));
  atomicMax(p + 2, __float_as_int(fmaxf(v.z, 0.f) + EPS_GEN));
  atomicMax(p + 3, __float_as_int(fmaxf(v.w, 0.f) + EPS_GEN));
}

// ------------------- edge pass 2: softmax numerator/denominator ------------
__global__ __launch_bounds__(256) void edge_sum_kernel(
    const float* __restrict__ t, const int* __restrict__ src,
    const int* __restrict__ dst, const float* __restrict__ maxb,
    float* __restrict__ denom, float* __restrict__ num) {
  const int e = blockIdx.x * 8 + (threadIdx.x >> 5);
  const int lane = threadIdx.x & 31;
  if (e >= EE) return;
  const int s = src[e];
  const int d = dst[e];
  const float4 v = *(const float4*)(t + (size_t)s * HH + lane * 4);
  const float4 m = *(const float4*)(maxb + (size_t)d * HH + lane * 4);
  float msg[4], a[4];
  msg[0] = fmaxf(v.x, 0.f) + EPS_GEN;
  msg[1] = fmaxf(v.y, 0.f) + EPS_GEN;
  msg[2] = fmaxf(v.z, 0.f) + EPS_GEN;
  msg[3] = fmaxf(v.w, 0.f) + EPS_GEN;
  a[0] = __expf(msg[0] - m.x);
  a[1] = __expf(msg[1] - m.y);
  a[2] = __expf(msg[2] - m.z);
  a[3] = __expf(msg[3] - m.w);
  float* pd = denom + (size_t)d * HH + lane * 4;
  float* pn = num + (size_t)d * HH + lane * 4;
#pragma unroll
  for (int i = 0; i < 4; ++i) {
    atomicAdd(pd + i, a[i]);
    atomicAdd(pn + i, a[i] * msg[i]);
  }
}

// --------------- combine: h2 = num/denom + t  (in place in num) ------------
__global__ __launch_bounds__(256) void combine_kernel(
    float* __restrict__ num, const float* __restrict__ denom,
    const float* __restrict__ t) {
  const int i = blockIdx.x * blockDim.x + threadIdx.x;
  if (i >= (NN * HH) / 4) return;
  const float4 n  = ((const float4*)num)[i];
  const float4 de = ((const float4*)denom)[i];
  const float4 tv = ((const float4*)t)[i];
  float4 o;
  o.x = n.x / (de.x > 0.f ? de.x : 1.f) + tv.x;
  o.y = n.y / (de.y > 0.f ? de.y : 1.f) + tv.y;
  o.z = n.z / (de.z > 0.f ? de.z : 1.f) + tv.z;
  o.w = n.w / (de.w > 0.f ? de.w : 1.f) + tv.w;
  ((float4*)num)[i] = o;
}

// -------------------- BatchNorm: column partial sums -----------------------
#define BN_ROWS 256
__global__ __launch_bounds__(256) void bn_reduce_kernel(
    const float* __restrict__ g1, float* __restrict__ sums, int nrows) {
  const int col = threadIdx.x;                 // 256 columns
  const int r0 = blockIdx.x * BN_ROWS;
  const int rend = min(r0 + BN_ROWS, nrows);
  float s = 0.f, sq = 0.f;
  for (int r = r0; r < rend; ++r) {
    const float v = g1[(size_t)r * HH2 + col];
    s += v;
    sq += v * v;
  }
  atomicAdd(&sums[col], s);
  atomicAdd(&sums[HH2 + col], sq);
}

__global__ void bn_finalize_kernel(const float* __restrict__ sums,
                                   const float* __restrict__ g,
                                   const float* __restrict__ b,
                                   float* __restrict__ ab) {
  const int c = threadIdx.x;
  const float inv_n = 1.f / (float)NN;
  const float mu = sums[c] * inv_n;
  const float var = sums[HH2 + c] * inv_n - mu * mu;
  const float a = g[c] * rsqrtf(var + EPS_NRM);
  ab[c] = a;
  ab[HH2 + c] = b[c] - mu * a;
}

__global__ __launch_bounds__(256) void bn_apply_relu_kernel(
    float* __restrict__ g1, const float* __restrict__ ab) {
  const int i = blockIdx.x * blockDim.x + threadIdx.x;
  if (i >= NN * HH2) return;
  const int col = i & (HH2 - 1);
  g1[i] = fmaxf(g1[i] * ab[col] + ab[HH2 + col], 0.f);
}

// ---------------------------------------------------------------------------
extern "C" void kernel_launch(void* const* d_in, const int* in_sizes, int n_in,
                              void* d_out, int out_size, void* d_ws, size_t ws_size,
                              hipStream_t stream) {
  const float* x     = (const float*)d_in[0];
  const int*   ei    = (const int*)d_in[1];
  const float* enc_w = (const float*)d_in[2];
  const float* enc_b = (const float*)d_in[3];
  const float* ln_g  = (const float*)d_in[4];
  const float* ln_b  = (const float*)d_in[5];
  const float* w1    = (const float*)d_in[6];
  const float* b1    = (const float*)d_in[7];
  const float* bn_g  = (const float*)d_in[8];
  const float* bn_b  = (const float*)d_in[9];
  const float* w2    = (const float*)d_in[10];
  const float* b2    = (const float*)d_in[11];
  const float* clf_w = (const float*)d_in[12];
  const float* clf_b = (const float*)d_in[13];
  float* out = (float*)d_out;

  const int* src = ei;
  const int* dst = ei + EE;

  // workspace partition (floats)
  float* ws    = (float*)d_ws;
  float* h     = ws;                       // [NN,HH]
  float* t     = h     + (size_t)NN * HH;  // [NN,HH]
  float* maxb  = t     + (size_t)NN * HH;  // [NN,HH] (int-punned fp32 bits)
  float* denom = maxb  + (size_t)NN * HH;  // [NN,HH]
  float* num   = denom + (size_t)NN * HH;  // [NN,HH] -> becomes h2
  float* g1    = num   + (size_t)NN * HH;  // [NN,HH2]
  float* sums  = g1    + (size_t)NN * HH2; // [2*HH2] sum/sumsq
  float* ab    = sums  + 2 * HH2;          // [2*HH2] scale/shift

  // 1) encoder: h = x @ enc_w + enc_b
  gemm_wmma_f32<0, 0, HH, HH><<<dim3(NN / 16, 1), 256, 0, stream>>>(
      x, enc_w, enc_b, h);

  for (int l = 0; l < NLAYERS; ++l) {
    // 2a) t = relu(LayerNorm(h))
    ln_relu_kernel<<<NN / 8, 256, 0, stream>>>(h, ln_g + l * HH, ln_b + l * HH, t);

    // 2b) clear segment buffers
    hipMemsetAsync(maxb, 0, (size_t)NN * HH * sizeof(float), stream);
    hipMemsetAsync(denom, 0, (size_t)NN * HH * sizeof(float), stream);
    hipMemsetAsync(num, 0, (size_t)NN * HH * sizeof(float), stream);
    hipMemsetAsync(sums, 0, (size_t)2 * HH2 * sizeof(float), stream);

    // 2c) softmax aggregation over incoming edges
    edge_max_kernel<<<EE / 8, 256, 0, stream>>>(t, src, dst, (int*)maxb);
    edge_sum_kernel<<<EE / 8, 256, 0, stream>>>(t, src, dst, maxb, denom, num);
    combine_kernel<<<(NN * HH / 4 + 255) / 256, 256, 0, stream>>>(num, denom, t);

    // 2d) MLP: g1 = h2 @ w1 + b1 ; BN ; relu ; h += g1 @ w2 + b2
    gemm_wmma_f32<0, 0, HH2, HH><<<dim3(NN / 16, HH2 / 128), 256, 0, stream>>>(
        num, w1 + (size_t)l * HH * HH2, b1 + l * HH2, g1);
    bn_reduce_kernel<<<(NN + BN_ROWS - 1) / BN_ROWS, 256, 0, stream>>>(g1, sums, NN);
    bn_finalize_kernel<<<1, HH2, 0, stream>>>(sums, bn_g + l * HH2, bn_b + l * HH2, ab);
    bn_apply_relu_kernel<<<(NN * HH2 + 255) / 256, 256, 0, stream>>>(g1, ab);
    gemm_wmma_f32<1, 0, HH, HH2><<<dim3(NN / 16, 1), 256, 0, stream>>>(
        g1, w2 + (size_t)l * HH2 * HH, b2 + l * HH, h);
  }

  // 3) classifier: out = h @ clf_w + clf_b   (N=64 -> waves 4..7 idle)
  gemm_wmma_f32<0, 0, OUTC, HH><<<dim3(NN / 16, 1), 256, 0, stream>>>(
      h, clf_w, clf_b, out);
}